// RGCNModel_74844100100818
// MI455X (gfx1250) — compile-verified
//
#include <hip/hip_runtime.h>

// ---------------- problem constants (from reference) ----------------
#define N_NODES 100002
#define DIM     128
#define N_REL   12
#define NBLK    4
#define BSZ     32          // DIM / NBLK
#define N_EDGE  1000000
#define BATCH   1024
#define N_ASP   8

#define W_ELEMS (N_REL * NBLK * BSZ * BSZ)   // 49152 floats = 192 KB

typedef float v2f __attribute__((ext_vector_type(2)));
typedef float v8f __attribute__((ext_vector_type(8)));

// ---------------- kernel 0: zero workspace (agg[N*D] + deg[N]) ----------------
__global__ void zero_ws_kernel(float* __restrict__ ws, long long n) {
    long long i = (long long)blockIdx.x * blockDim.x + threadIdx.x;
    long long stride = (long long)gridDim.x * blockDim.x;
    for (; i < n; i += stride) ws[i] = 0.0f;
}

// ---------------- kernel 1: per-edge block-diag transform + scatter-add -------
// One wave (32 lanes) per edge; lane = output element within a 32-wide block.
// All relation weights live in LDS (192 KB of the WGP's 320 KB).
__global__ void edge_scatter_kernel(const float* __restrict__ node_feats,
                                    const float* __restrict__ weight,
                                    const int*   __restrict__ edge_index,  // [2,E]
                                    const int*   __restrict__ edge_type,   // [E]
                                    const float* __restrict__ edge_weight, // [E]
                                    float* __restrict__ agg,               // [N,DIM]
                                    float* __restrict__ deg)               // [N]
{
    extern __shared__ float sW[];   // W_ELEMS floats
    for (int i = threadIdx.x; i < W_ELEMS; i += blockDim.x) sW[i] = weight[i];
    __syncthreads();

    const int lane          = threadIdx.x & 31;
    const int wave          = threadIdx.x >> 5;
    const int wavesPerBlock = blockDim.x >> 5;
    const int waveStride    = gridDim.x * wavesPerBlock;

    for (int e = blockIdx.x * wavesPerBlock + wave; e < N_EDGE; e += waveStride) {
        const int   dst = edge_index[e];            // agg node  (edge_index[0][e])
        const int   src = edge_index[N_EDGE + e];   // gather    (edge_index[1][e])
        const int   rel = edge_type[e];
        const float ew  = edge_weight[e];

        const float* __restrict__ xrow = node_feats + (long long)src * DIM;
        const float* __restrict__ wr   = sW + rel * (NBLK * BSZ * BSZ);
        float*       __restrict__ orow = agg + (long long)dst * DIM;

#pragma unroll
        for (int b = 0; b < NBLK; ++b) {
            // coalesced 128B gather of this block's inputs, scaled by edge weight
            const float xv = xrow[b * BSZ + lane] * ew;
            const float* __restrict__ wb = wr + b * BSZ * BSZ;
            float acc = 0.0f;
#pragma unroll
            for (int i = 0; i < BSZ; ++i) {
                // broadcast x_i to all lanes via v_readlane (immediate lane idx)
                float xi = __builtin_bit_cast(
                    float, __builtin_amdgcn_readlane(__builtin_bit_cast(int, xv), i));
                acc = fmaf(xi, wb[i * BSZ + lane], acc);  // LDS read, conflict-free
            }
            atomicAdd(orow + b * BSZ + lane, acc);        // global_atomic_add_f32
        }
        if (lane == 0) atomicAdd(deg + dst, 1.0f);
    }
}

// ---------------- kernel 2: fused  feats = relu(agg/deg + A@root + bias) ------
// WMMA f32 16x16x4: one wave per 16x16 tile; 8 waves/block cover all DIM columns
// of one 16-row stripe.  Epilogue fused; result written in place over `agg`.
__global__ void __launch_bounds__(256)
root_gemm_fused_kernel(const float* __restrict__ node_feats,  // A [N,DIM]
                       const float* __restrict__ root,        // B [DIM,DIM]
                       const float* __restrict__ bias,        // [DIM]
                       const float* __restrict__ deg,         // [N]
                       float*       __restrict__ agg)         // in/out [N,DIM]
{
    const int lane    = threadIdx.x & 31;
    const int wave    = threadIdx.x >> 5;       // column tile 0..7
    const int lane15  = lane & 15;
    const int halfSel = lane >> 4;              // 0 or 1 (K / M half select)

    const int rowBase = blockIdx.x * 16;
    const int col     = wave * 16 + lane15;

    // A fragment row for this lane (clamped: WMMA requires EXEC all-1s; bogus
    // rows produce bogus C rows that the predicated store below discards)
    int rowA = rowBase + lane15;
    if (rowA >= N_NODES) rowA = N_NODES - 1;
    const float* __restrict__ arow = node_feats + (long long)rowA * DIM;

    v8f c = {};
#pragma unroll
    for (int kk = 0; kk < DIM; kk += 4) {
        const int ka = kk + halfSel * 2;                     // ISA 16x4 A layout
        v2f a = *(const v2f*)(arow + ka);                    // K = ka, ka+1
        v2f b;
        b.x = root[(long long)ka * DIM + col];               // mirrored 4x16 B layout
        b.y = root[(long long)(ka + 1) * DIM + col];
        c = __builtin_amdgcn_wmma_f32_16x16x4_f32(
                /*neg_a=*/false, a, /*neg_b=*/false, b,
                /*c_mod=*/(short)0, c, /*reuse_a=*/false, /*reuse_b=*/false);
    }

    const float bb = bias[col];
#pragma unroll
    for (int v = 0; v < 8; ++v) {
        const int row = rowBase + halfSel * 8 + v;           // C layout: VGPR v -> M=v / v+8
        if (row < N_NODES) {
            float dg = deg[row];
            dg = dg > 1.0f ? dg : 1.0f;                      // mean aggr: max(deg,1)
            const long long idx = (long long)row * DIM + col;
            float val = agg[idx] / dg + c[v] + bb;
            agg[idx] = val > 0.0f ? val : 0.0f;              // ReLU
        }
    }
}

// ---------------- kernel 3: gather user/item/aspect reps ----------------------
__global__ void gather_kernel(const float* __restrict__ feats,
                              const int* __restrict__ user_ids,    // [B]
                              const int* __restrict__ item_ids,    // [B]
                              const int* __restrict__ aspect_ids,  // [B,A]
                              float* __restrict__ out)             // concat flat
{
    const int total = BATCH * DIM * 2 + BATCH * N_ASP * DIM;
    int tid = blockIdx.x * blockDim.x + threadIdx.x;
    if (tid >= total) return;

    int row;
    if (tid < BATCH * DIM) {                       // ent_user_rep [B,1,D]
        row = user_ids[tid / DIM];
    } else if (tid < 2 * BATCH * DIM) {            // ent_item_rep [B,1,D]
        int t = tid - BATCH * DIM;
        row = item_ids[t / DIM];
    } else {                                       // ent_aspect_rep [B,A,D]
        int t = tid - 2 * BATCH * DIM;
        row = aspect_ids[t / DIM];                 // flat [B*A] index
    }
    const int d = tid % DIM;
    out[tid] = feats[(long long)row * DIM + d];
}

// ---------------- host-side launch --------------------------------------------
extern "C" void kernel_launch(void* const* d_in, const int* in_sizes, int n_in,
                              void* d_out, int out_size, void* d_ws, size_t ws_size,
                              hipStream_t stream) {
    const float* node_feats  = (const float*)d_in[0];
    const float* weight      = (const float*)d_in[1];
    const float* root        = (const float*)d_in[2];
    const float* bias        = (const float*)d_in[3];
    const int*   edge_index  = (const int*)  d_in[4];
    const int*   edge_type   = (const int*)  d_in[5];
    const float* edge_weight = (const float*)d_in[6];
    const int*   user_ids    = (const int*)  d_in[7];
    const int*   item_ids    = (const int*)  d_in[8];
    const int*   aspect_ids  = (const int*)  d_in[9];
    float*       out         = (float*)d_out;

    float* agg = (float*)d_ws;                         // [N_NODES, DIM]
    float* deg = agg + (long long)N_NODES * DIM;       // [N_NODES]

    // 0) zero agg + deg
    const long long zn = (long long)N_NODES * DIM + N_NODES;
    zero_ws_kernel<<<2048, 256, 0, stream>>>(agg, zn);

    // 1) edge scatter: 192 KB dynamic LDS per block (fits WGP's 320 KB)
    const size_t wBytes = (size_t)W_ELEMS * sizeof(float);
    (void)hipFuncSetAttribute((const void*)edge_scatter_kernel,
                              hipFuncAttributeMaxDynamicSharedMemorySize,
                              (int)wBytes);
    edge_scatter_kernel<<<3072, 256, wBytes, stream>>>(
        node_feats, weight, edge_index, edge_type, edge_weight, agg, deg);

    // 2) fused root GEMM (WMMA f32) + mean + bias + ReLU, in place over agg
    const int mtiles = (N_NODES + 15) / 16;            // 6251
    root_gemm_fused_kernel<<<mtiles, 256, 0, stream>>>(
        node_feats, root, bias, deg, agg);

    // 3) gather outputs (user | item | aspect, concatenated flat)
    const int total = BATCH * DIM * 2 + BATCH * N_ASP * DIM;
    gather_kernel<<<(total + 255) / 256, 256, 0, stream>>>(
        agg, user_ids, item_ids, aspect_ids, out);
}